// pointLoss_13013750906955
// MI455X (gfx1250) — compile-verified
//
#include <hip/hip_runtime.h>

// Problem constants (from reference)
#define R2      10            // RADIUS // 2
#define WIN_I   21            // 2*R2 + 1   (di in [-10, 10])
#define WIN_J   20            // 2*R2       (dj in [-10, 9])
#define WIN     (WIN_I * WIN_J)   // 420 window elements per point
#define NC      3
#define B_      16
#define L_      100
#define H_      1024
#define W_      1024
#define HW      (H_ * W_)
#define NBLOCKS (B_ * L_)     // 1600 points -> 1600 workgroups

#define CE_C0     1.5514447139320509f        // log(e + 2)
#define GAUSS_K  -0.010204081632653061f      // -0.5 / (7*7)
#define INV_N     5.9604644775390625e-8f     // 1 / (16*1024*1024) == 2^-24

typedef __attribute__((ext_vector_type(2))) float v2f;
typedef __attribute__((ext_vector_type(8))) float v8f;

// Kernel 1: one workgroup per point. Sparse evaluation of mask * ce over the
// 21x20 window only; everything else in the image contributes exactly zero.
__global__ __launch_bounds__(256) void point_ce_partial(
    const float* __restrict__ y_pred,        // (B, 3, H, W) f32
    const int*   __restrict__ y_true,        // (B, 1, H, W) i32
    const int*   __restrict__ points,        // (B, L, 2)    i32
    const int*   __restrict__ point_labels,  // (B, L, 1)    i32
    float*       __restrict__ partial)       // (1600,) block sums
{
  const int blk   = blockIdx.x;              // b*L + l  (uniform -> scalar loads)
  const int b     = blk / L_;
  const int pi    = points[blk * 2 + 0];
  const int pj    = points[blk * 2 + 1];
  const int label = point_labels[blk];
  const int tid   = threadIdx.x;

  const int*   yt_b = y_true + (size_t)b * HW;
  const float* yp_b = y_pred + (size_t)b * NC * HW;

  float acc = 0.0f;
  for (int w = tid; w < WIN; w += 256) {     // <= 2 iterations
    const int i  = w / WIN_J;
    const int j  = w - i * WIN_J;
    const int dI = i - R2;                   // [-10, 10]
    const int dJ = j - R2;                   // [-10,  9]
    const int ii = pi + dI;
    const int jj = pj + dJ;
    if ((unsigned)ii < (unsigned)H_ && (unsigned)jj < (unsigned)W_) {
      const int off = ii * W_ + jj;
      const int t   = yt_b[off];
      if (t == label) {
        const float x0 = yp_b[off];
        const float x1 = yp_b[off + HW];
        const float x2 = yp_b[off + 2 * HW];
        const float m  = fmaxf(x0, fmaxf(x1, x2));
        const float e0 = __expf(x0 - m);
        const float e1 = __expf(x1 - m);
        const float e2 = __expf(x2 - m);
        const float et = (t == 0) ? e0 : ((t == 1) ? e1 : e2);
        const float p  = et / (e0 + e1 + e2);                    // softmax[true class]
        const float g  = __expf(GAUSS_K * (float)(dI * dI + dJ * dJ));
        acc += g * (CE_C0 - p);              // gauss * (log(e+2) - p_true)
      }
    }
  }

  // Deterministic block tree reduction.
  __shared__ float red[256];
  red[tid] = acc;
  __syncthreads();
  for (int s = 128; s > 0; s >>= 1) {
    if (tid < s) red[tid] += red[tid + s];
    __syncthreads();
  }
  if (tid == 0) partial[blk] = red[0];
}

// Kernel 2: deterministic final reduction of the 1600 partials. The last
// 32 -> 1 step runs on the matrix pipe: D = A(16x4) * ones(4x16) + 0, with
// lane values placed in A's K=0 / K=2 slots, so every D row = v_m + v_{m+16}.
__global__ __launch_bounds__(256) void point_ce_final(
    const float* __restrict__ partial, float* __restrict__ out)
{
  const int tid = threadIdx.x;
  float s = 0.0f;
  for (int i = tid; i < NBLOCKS; i += 256) s += partial[i];   // fixed order

  __shared__ float red[256];
  red[tid] = s;
  __syncthreads();
  for (int k = 128; k >= 32; k >>= 1) {
    if (tid < k) red[tid] += red[tid + k];
    __syncthreads();
  }

  // red[0..31] now holds 32 values. Every wave performs the identical WMMA
  // (EXEC all-ones, no divergence); only thread 0 writes the result.
  const float v = red[tid & 31];
  v2f a;  a.x = v;    a.y = 0.0f;   // A 16x4: VGPR0 lanes0-15 -> K=0, lanes16-31 -> K=2
  v2f bb; bb.x = 1.0f; bb.y = 1.0f; // B 4x16: all ones
  v8f c = {};
  // D[m][n] = v_m + v_{m+16} for all n.
  v8f d = __builtin_amdgcn_wmma_f32_16x16x4_f32(
      /*neg_a=*/false, a, /*neg_b=*/false, bb,
      /*c_mod=*/(short)0, c, /*reuse_a=*/false, /*reuse_b=*/false);
  // Per lane: lanes 0-15 hold rows 0..7 (col n), lanes 16-31 hold rows 8..15.
  float t = d[0] + d[1] + d[2] + d[3] + d[4] + d[5] + d[6] + d[7];
  t += __shfl_xor(t, 16, 32);       // combine the two half-wave row groups
  if (tid == 0) out[0] = t * INV_N;
}

extern "C" void kernel_launch(void* const* d_in, const int* in_sizes, int n_in,
                              void* d_out, int out_size, void* d_ws, size_t ws_size,
                              hipStream_t stream)
{
  const float* y_pred       = (const float*)d_in[0];
  const int*   y_true       = (const int*)d_in[1];
  const int*   points       = (const int*)d_in[2];
  const int*   point_labels = (const int*)d_in[3];
  float*       partial      = (float*)d_ws;    // 1600 floats, written every call
  float*       out          = (float*)d_out;

  point_ce_partial<<<NBLOCKS, 256, 0, stream>>>(y_pred, y_true, points,
                                                point_labels, partial);
  point_ce_final<<<1, 256, 0, stream>>>(partial, out);
}